// QCDAware_70884140253333
// MI455X (gfx1250) — compile-verified
//
#include <hip/hip_runtime.h>
#include <math.h>

// QCDAware forward for MI455X (gfx1250, wave32).
// B=32, F=6, N=1024. Output 128MB f32 -> HBM-store floor ~5.5us @ 23.3TB/s.
// emb lives in LDS padded to 8 feature rows (rows 6,7 = 0) so both WMMA
// fragments load branchlessly; Gram tiles via V_WMMA_F32_16X16X4_F32 (two K=4
// WMMAs cover K=6); softmax uses one v_exp_f32 per element; diagonal fix-up is
// a wave-uniform (readfirstlane -> SALU) branch on a single tile per strip.

typedef float v2f __attribute__((ext_vector_type(2)));
typedef float v8f __attribute__((ext_vector_type(8)));

#define NB 32
#define NF 6
#define NFP 8                          // padded feature rows (6,7 zeroed)
#define NN 1024
#define NTILE (NN / 16)                // 64 j-tiles per row strip
#define WAVES 4                        // waves per block
#define BTHREADS (WAVES * 32)          // 128 threads
#define BLK_PER_B (NN / (16 * WAVES))  // 16 blocks per batch (one 16-row strip per wave)

#define LOG2E 1.44269504088896340736f

// ---- monotonic float <-> uint mapping (deterministic atomic min/max) ----
__device__ __forceinline__ unsigned f2o(float f) {
  unsigned u = __float_as_uint(f);
  return (u & 0x80000000u) ? ~u : (u | 0x80000000u);
}
__device__ __forceinline__ float o2f(unsigned o) {
  unsigned u = (o & 0x80000000u) ? (o & 0x7FFFFFFFu) : ~o;
  return __uint_as_float(u);
}

// ---- 16x16 gram tile: C = A(16x8) * B(8x16) via two 16x16x4 f32 WMMAs ----
__device__ __forceinline__ v8f gram2(v2f a1, v2f a2, v2f b1, v2f b2) {
  v8f c = {};
  c = __builtin_amdgcn_wmma_f32_16x16x4_f32(false, a1, false, b1, (short)0, c, false, false);
  c = __builtin_amdgcn_wmma_f32_16x16x4_f32(false, a2, false, b2, (short)0, c, false, false);
  return c;
}

// ---- cooperative load: emb batch -> LDS (zero-padded), sq/pm interleaved ----
__device__ __forceinline__ void load_batch(const float* __restrict__ embB, float alpha,
                                           float* sE, float* sQP) {
  for (int t = threadIdx.x; t < NF * NN; t += BTHREADS) sE[t] = embB[t];
  for (int t = NF * NN + threadIdx.x; t < NFP * NN; t += BTHREADS) sE[t] = 0.0f;
  __syncthreads();
  for (int n = threadIdx.x; n < NN; n += BTHREADS) {
    float s = 0.0f;
#pragma unroll
    for (int f = 0; f < NF; ++f) { float v = sE[f * NN + n]; s += v * v; }
    sQP[2 * n + 0] = s;                              // |x_n|^2
    float mm = sE[4 * NN + n];                       // momentum row
    sQP[2 * n + 1] = expf(2.0f * alpha * logf(mm));  // m^(2*alpha), accurate
  }
  __syncthreads();
}

// ---- raw d_ij tile values (8 rows per lane), no diagonal handling ----
__device__ __forceinline__ void tile_dv(const float* __restrict__ sE,
                                        const float* __restrict__ sQP,
                                        v2f a1, v2f a2, int hi, int m, int jt,
                                        const float* sqi, const float* pmi,
                                        float dv[8]) {
  const int j = jt * 16 + m;
  v2f b1, b2;
  b1.x = sE[(2 * hi + 0) * NN + j];
  b1.y = sE[(2 * hi + 1) * NN + j];
  b2.x = sE[(2 * hi + 4) * NN + j];   // hi=1 hits zero-padded rows 6,7
  b2.y = sE[(2 * hi + 5) * NN + j];
  v8f c = gram2(a1, a2, b1, b2);
  const float sqj = sQP[2 * j + 0];
  const float pmj = sQP[2 * j + 1];
#pragma unroll
  for (int v = 0; v < 8; ++v) {
    float sd = fmaf(c[v], -2.0f, sqi[v] + sqj);   // sq_i + sq_j - 2*g_ij
    dv[v] = sd * fminf(pmi[v], pmj);
  }
}

// ================= kernel 1: global max / per-batch off-diag min =================
__global__ __launch_bounds__(BTHREADS) void qcd_reduce(const float* __restrict__ emb,
                                                       const float* __restrict__ alphaP,
                                                       unsigned* __restrict__ ws) {
  __shared__ float sE[NFP * NN];
  __shared__ float sQP[2 * NN];
  const int b = blockIdx.x / BLK_PER_B;
  const int sg = blockIdx.x % BLK_PER_B;
  load_batch(emb + (size_t)b * NF * NN, alphaP[0], sE, sQP);

  const int lane = threadIdx.x & 31;
  const int wave = threadIdx.x >> 5;
  const int i0 = __builtin_amdgcn_readfirstlane((sg * WAVES + wave) * 16);
  const int jdiag = i0 >> 4;            // wave-uniform: tile containing the diagonal
  const int m = lane & 15;
  const int hi = lane >> 4;
  const int vd = m - 8 * hi;            // this lane's diagonal row index (if 0..7)

  v2f a1, a2;
  a1.x = sE[(2 * hi + 0) * NN + i0 + m];
  a1.y = sE[(2 * hi + 1) * NN + i0 + m];
  a2.x = sE[(2 * hi + 4) * NN + i0 + m];
  a2.y = sE[(2 * hi + 5) * NN + i0 + m];

  float sqi[8], pmi[8];
#pragma unroll
  for (int v = 0; v < 8; ++v) {
    const int iv = i0 + v + 8 * hi;
    sqi[v] = sQP[2 * iv + 0];
    pmi[v] = sQP[2 * iv + 1];
  }

  float vmax = -__builtin_inff();
  float vmin = __builtin_inff();
  float dv[8];
  for (int jt = 0; jt < NTILE; ++jt) {
    tile_dv(sE, sQP, a1, a2, hi, m, jt, sqi, pmi, dv);
    if (jt == jdiag) {                  // scalar branch (jdiag is SGPR)
#pragma unroll
      for (int v = 0; v < 8; ++v) {
        vmax = fmaxf(vmax, (v == vd) ? 0.0f : dv[v]);             // ref diag d_ij == 0
        vmin = fminf(vmin, (v == vd) ? __builtin_inff() : dv[v]); // min excludes diag
      }
    } else {
#pragma unroll
      for (int v = 0; v < 8; ++v) {
        vmax = fmaxf(vmax, dv[v]);
        vmin = fminf(vmin, dv[v]);
      }
    }
  }
#pragma unroll
  for (int off = 16; off >= 1; off >>= 1) {
    vmax = fmaxf(vmax, __shfl_xor(vmax, off, 32));
    vmin = fminf(vmin, __shfl_xor(vmin, off, 32));
  }
  if (lane == 0) {
    atomicMax(&ws[0], f2o(vmax));      // global max over all batches
    atomicMin(&ws[1 + b], f2o(vmin));  // per-batch min
  }
}

// ================= kernel 2: row softmax, single 128MB write =================
__global__ __launch_bounds__(BTHREADS) void qcd_softmax(const float* __restrict__ emb,
                                                        const float* __restrict__ alphaP,
                                                        const float* __restrict__ betaP,
                                                        const unsigned* __restrict__ ws,
                                                        float* __restrict__ out) {
  __shared__ float sE[NFP * NN];
  __shared__ float sQP[2 * NN];
  const int b = blockIdx.x / BLK_PER_B;
  const int sg = blockIdx.x % BLK_PER_B;
  load_batch(emb + (size_t)b * NF * NN, alphaP[0], sE, sQP);

  const float beta = betaP[0];
  const float big = 1000.0f * o2f(ws[0]);
  const float dmin = o2f(ws[1 + b]);
  // x - rowmax(x) = -beta^2/dmin * (dv - rowmin(dv));  e = 2^(kk*dv + base)
  const float kk = -(beta * beta) * LOG2E / dmin;

  const int lane = threadIdx.x & 31;
  const int wave = threadIdx.x >> 5;
  const int i0 = __builtin_amdgcn_readfirstlane((sg * WAVES + wave) * 16);
  const int jdiag = i0 >> 4;
  const int m = lane & 15;
  const int hi = lane >> 4;
  const int vd = m - 8 * hi;

  v2f a1, a2;
  a1.x = sE[(2 * hi + 0) * NN + i0 + m];
  a1.y = sE[(2 * hi + 1) * NN + i0 + m];
  a2.x = sE[(2 * hi + 4) * NN + i0 + m];
  a2.y = sE[(2 * hi + 5) * NN + i0 + m];

  float sqi[8], pmi[8];
#pragma unroll
  for (int v = 0; v < 8; ++v) {
    const int iv = i0 + v + 8 * hi;
    sqi[v] = sQP[2 * iv + 0];
    pmi[v] = sQP[2 * iv + 1];
  }

  float dv[8];

  // --- pass 1: per-row min of dv (== row max of d_norm; no exp/scale needed) ---
  float rmin[8];
#pragma unroll
  for (int v = 0; v < 8; ++v) rmin[v] = __builtin_inff();
  for (int jt = 0; jt < NTILE; ++jt) {
    tile_dv(sE, sQP, a1, a2, hi, m, jt, sqi, pmi, dv);
    if (jt == jdiag) {
#pragma unroll
      for (int v = 0; v < 8; ++v) rmin[v] = fminf(rmin[v], (v == vd) ? big : dv[v]);
    } else {
#pragma unroll
      for (int v = 0; v < 8; ++v) rmin[v] = fminf(rmin[v], dv[v]);
    }
  }
#pragma unroll
  for (int off = 8; off >= 1; off >>= 1) {   // reduce within each 16-lane half
#pragma unroll
    for (int v = 0; v < 8; ++v) rmin[v] = fminf(rmin[v], __shfl_xor(rmin[v], off, 32));
  }
  float base[8];
#pragma unroll
  for (int v = 0; v < 8; ++v) base[v] = -kk * rmin[v];

  // --- pass 2: row sums of e = 2^(kk*dv + base) ---
  float srow[8];
#pragma unroll
  for (int v = 0; v < 8; ++v) srow[v] = 0.0f;
  for (int jt = 0; jt < NTILE; ++jt) {
    tile_dv(sE, sQP, a1, a2, hi, m, jt, sqi, pmi, dv);
    if (jt == jdiag) {
#pragma unroll
      for (int v = 0; v < 8; ++v) if (v == vd) dv[v] = big;
    }
#pragma unroll
    for (int v = 0; v < 8; ++v)
      srow[v] += __builtin_amdgcn_exp2f(fmaf(kk, dv[v], base[v]));
  }
#pragma unroll
  for (int off = 8; off >= 1; off >>= 1) {
#pragma unroll
    for (int v = 0; v < 8; ++v) srow[v] += __shfl_xor(srow[v], off, 32);
  }
  float rs[8];
#pragma unroll
  for (int v = 0; v < 8; ++v) rs[v] = 1.0f / srow[v];

  // --- pass 3: write normalized softmax (the only HBM-heavy step) ---
  for (int jt = 0; jt < NTILE; ++jt) {
    tile_dv(sE, sQP, a1, a2, hi, m, jt, sqi, pmi, dv);
    if (jt == jdiag) {
#pragma unroll
      for (int v = 0; v < 8; ++v) if (v == vd) dv[v] = big;
    }
    const int j = jt * 16 + m;
#pragma unroll
    for (int v = 0; v < 8; ++v) {
      const int iv = i0 + v + 8 * hi;
      out[((size_t)(b * NN + iv)) * NN + j] =
          __builtin_amdgcn_exp2f(fmaf(kk, dv[v], base[v])) * rs[v];
    }
  }
}

// ================= workspace init (re-run every call; graph-safe) =================
__global__ void qcd_init_ws(unsigned* __restrict__ ws) {
  const int t = threadIdx.x;
  if (t == 0) ws[0] = f2o(-__builtin_inff());
  if (t >= 1 && t < 1 + NB) ws[t] = f2o(__builtin_inff());
}

extern "C" void kernel_launch(void* const* d_in, const int* in_sizes, int n_in,
                              void* d_out, int out_size, void* d_ws, size_t ws_size,
                              hipStream_t stream) {
  const float* emb   = (const float*)d_in[0];
  const float* alpha = (const float*)d_in[1];
  const float* beta  = (const float*)d_in[2];
  float* out = (float*)d_out;
  unsigned* ws = (unsigned*)d_ws;

  qcd_init_ws<<<1, 64, 0, stream>>>(ws);
  dim3 grid(NB * BLK_PER_B);  // 512 blocks, 4 waves each, one 16-row strip per wave
  qcd_reduce<<<grid, BTHREADS, 0, stream>>>(emb, alpha, ws);
  qcd_softmax<<<grid, BTHREADS, 0, stream>>>(emb, alpha, beta, ws, out);
}